// Quantiser_89739046683455
// MI455X (gfx1250) — compile-verified
//
#include <hip/hip_runtime.h>
#include <cstdint>

typedef __attribute__((ext_vector_type(2))) float v2f;
typedef __attribute__((ext_vector_type(8))) float v8f;

#define NTOK   65536      // 32*2048 tokens
#define DIM    256
#define VOCAB  4096
#define MB     128        // tokens per workgroup (resident in LDS)
#define NC     64         // codes per streamed chunk
#define NCHUNK (VOCAB / NC)
#define PAD    260        // stride%64==4 -> conflict-free b64 frags; stride*4%16==0 -> b128 ok
#define XS_F   (MB * PAD)
#define ES_F   (NC * PAD)
#define SMEM_BYTES ((XS_F + 2 * ES_F) * 4)   // 266,240 B <= 320 KB WGP LDS

// ---- gfx1250 async global->LDS copies (ASYNCcnt-tracked) ----
__device__ __forceinline__ void async_ld_b128(uint32_t lds_byte_off, const void* gaddr) {
  asm volatile("global_load_async_to_lds_b128 %0, %1, off"
               :: "v"(lds_byte_off), "v"(gaddr) : "memory");
}
__device__ __forceinline__ void wait_async_16() {
  asm volatile("s_wait_asynccnt 16" ::: "memory");
}
__device__ __forceinline__ void wait_async_0() {
  asm volatile("s_wait_asynccnt 0" ::: "memory");
}

// ---- kernel 1: e2[v] = ||emb[v]||^2 into ws[0..VOCAB) ----
__global__ __launch_bounds__(256) void vq_prep(const float* __restrict__ emb,
                                               float* __restrict__ ws) {
  int v = blockIdx.x * 256 + threadIdx.x;
  if (v < VOCAB) {
    const float4* row = (const float4*)(emb + (size_t)v * DIM);
    float s = 0.f;
#pragma unroll 8
    for (int i = 0; i < DIM / 4; ++i) {
      float4 q = row[i];
      s += q.x * q.x + q.y * q.y + q.z * q.z + q.w * q.w;
    }
    ws[v] = s;
  }
}

// ---- kernel 2: fused distance GEMM (f32 WMMA) + argmin + gather + loss partial ----
__global__ __launch_bounds__(256) void vq_main(const float* __restrict__ x,
                                               const float* __restrict__ emb,
                                               const float* __restrict__ e2g,
                                               float* __restrict__ out,
                                               float* __restrict__ partials) {
  extern __shared__ float smem[];
  float* xs = smem;            // MB x PAD
  float* es = smem + XS_F;     // 2 x (NC x PAD), double-buffered

  const int tid  = threadIdx.x;
  const int bid  = blockIdx.x;
  const int lane = tid & 31;
  const int w    = tid >> 5;       // wave id 0..7
  const int ln   = lane & 15;      // matrix row/col within fragment
  const int hi   = lane >> 4;      // K-subcolumn selector for 16x4 f32 layout
  const long tokBase = (long)bid * MB;

  // ---- stage X tile (MB x 256 f32) asynchronously: 32 b128 ops per thread ----
#pragma unroll 4
  for (int j = 0; j < 32; ++j) {
    int e   = tid + 256 * j;                 // 0..8191 (128 rows * 64 float4 units)
    int row = e >> 6;
    int c4  = e & 63;
    uint32_t loff = (uint32_t)((row * PAD + c4 * 4) * 4);
    async_ld_b128(loff, x + (tokBase + row) * DIM + c4 * 4);
  }

  auto stage_chunk = [&](int c, int buf) {
#pragma unroll 4
    for (int j = 0; j < 16; ++j) {
      int e   = tid + 256 * j;               // 0..4095 (64 rows * 64 float4 units)
      int row = e >> 6;
      int c4  = e & 63;
      uint32_t loff = (uint32_t)((XS_F + buf * ES_F + row * PAD + c4 * 4) * 4);
      async_ld_b128(loff, emb + (size_t)(c * NC + row) * DIM + c4 * 4);
    }
  };
  stage_chunk(0, 0);

  float bestD[8];
  int   bestI[8];
#pragma unroll
  for (int r = 0; r < 8; ++r) { bestD[r] = 3.4e38f; bestI[r] = 0; }

  const float* aBase = xs + (16 * w + ln) * PAD + 2 * hi;

  for (int c = 0; c < NCHUNK; ++c) {
    const int buf = c & 1;
    if (c + 1 < NCHUNK) { stage_chunk(c + 1, buf ^ 1); wait_async_16(); }
    else                { wait_async_0(); }
    __syncthreads();   // chunk c (and X on first iter) visible to all waves

    const float* b0 = es + buf * ES_F + ln * PAD + 2 * hi;
    v8f acc[4] = {v8f{}, v8f{}, v8f{}, v8f{}};

    // software-pipelined K loop: fragments for step kk+1 are fetched before
    // the WMMAs of step kk, so each wmma waits on loads issued a stage earlier
    v2f a  = *(const v2f*)(aBase);
    v2f f0 = *(const v2f*)(b0 + 0 * 16 * PAD);
    v2f f1 = *(const v2f*)(b0 + 1 * 16 * PAD);
    v2f f2 = *(const v2f*)(b0 + 2 * 16 * PAD);
    v2f f3 = *(const v2f*)(b0 + 3 * 16 * PAD);
#pragma unroll 8
    for (int kk = 0; kk < 64; ++kk) {         // K = 256, 4 per WMMA
      const int kn = (kk + 1) & 63;           // last prefetch wraps (harmless, no OOB)
      v2f an = *(const v2f*)(aBase + 4 * kn);
      v2f g0 = *(const v2f*)(b0 + 0 * 16 * PAD + 4 * kn);
      v2f g1 = *(const v2f*)(b0 + 1 * 16 * PAD + 4 * kn);
      v2f g2 = *(const v2f*)(b0 + 2 * 16 * PAD + 4 * kn);
      v2f g3 = *(const v2f*)(b0 + 3 * 16 * PAD + 4 * kn);
      acc[0] = __builtin_amdgcn_wmma_f32_16x16x4_f32(false, a, false, f0, (short)0, acc[0], false, false);
      acc[1] = __builtin_amdgcn_wmma_f32_16x16x4_f32(false, a, false, f1, (short)0, acc[1], false, false);
      acc[2] = __builtin_amdgcn_wmma_f32_16x16x4_f32(false, a, false, f2, (short)0, acc[2], false, false);
      acc[3] = __builtin_amdgcn_wmma_f32_16x16x4_f32(false, a, false, f3, (short)0, acc[3], false, false);
      a = an; f0 = g0; f1 = g1; f2 = g2; f3 = g3;
    }

    const int v0 = c * NC;
#pragma unroll
    for (int t = 0; t < 4; ++t) {
      const int   vidx = v0 + 16 * t + ln;
      const float e2v  = e2g[vidx];
#pragma unroll
      for (int r = 0; r < 8; ++r) {           // VGPR r holds rows r / r+8 (by lane group)
        float d = fmaf(-2.f, acc[t][r], e2v); // ||x||^2 common -> dropped
        if (d < bestD[r]) { bestD[r] = d; bestI[r] = vidx; }
      }
    }
    __syncthreads();   // all waves done reading es[buf] before it is overwritten
  }

  // ---- cross-lane argmin over the 16 lanes that share each row ----
#pragma unroll
  for (int r = 0; r < 8; ++r) {
#pragma unroll
    for (int m = 8; m >= 1; m >>= 1) {
      float dO = __shfl_xor(bestD[r], m, 32);
      int   iO = __shfl_xor(bestI[r], m, 32);
      if (dO < bestD[r] || (dO == bestD[r] && iO < bestI[r])) { bestD[r] = dO; bestI[r] = iO; }
    }
  }

  // ---- gather emb[idx] -> out, accumulate loss partial ----
  float lsum = 0.f;
#pragma unroll 2
  for (int mi = 0; mi < 16; ++mi) {
    const int r = mi & 7, g = mi >> 3;
    const int idxm = __shfl(bestI[r], g * 16, 32);
    const float* erow = emb + (size_t)idxm * DIM;
    const float* xrow = xs + (16 * w + mi) * PAD;
    float* orow = out + (tokBase + 16 * w + mi) * DIM;
#pragma unroll
    for (int h = 0; h < 2; ++h) {
      const int off = lane * 8 + h * 4;
      float4 ev = *(const float4*)(erow + off);
      float4 xv = *(const float4*)(xrow + off);   // 16B-aligned (PAD*4 % 16 == 0)
      *(float4*)(orow + off) = ev;
      float d0 = ev.x - xv.x, d1 = ev.y - xv.y, d2 = ev.z - xv.z, d3 = ev.w - xv.w;
      lsum += d0 * d0 + d1 * d1 + d2 * d2 + d3 * d3;
    }
  }
#pragma unroll
  for (int m = 1; m < 32; m <<= 1) lsum += __shfl_xor(lsum, m, 32);
  if (lane == 0) es[w] = lsum;               // es no longer needed; reuse as scratch
  __syncthreads();
  if (tid == 0) {
    float s = 0.f;
#pragma unroll
    for (int i = 0; i < 8; ++i) s += es[i];
    partials[bid] = s;                        // overwritten every launch: deterministic
  }
}

// ---- kernel 3: deterministic serial reduction of 512 partials -> loss scalar ----
__global__ void vq_final(const float* __restrict__ ws, float* __restrict__ out) {
  if (threadIdx.x == 0 && blockIdx.x == 0) {
    float s = 0.f;
    for (int i = 0; i < NTOK / MB; ++i) s += ws[VOCAB + i];
    // loss = codebook + 0.25*commit = 1.25 * mean((q - x)^2)
    out[(size_t)NTOK * DIM] = 1.25f * s / (float)((size_t)NTOK * DIM);
  }
}

extern "C" void kernel_launch(void* const* d_in, const int* in_sizes, int n_in,
                              void* d_out, int out_size, void* d_ws, size_t ws_size,
                              hipStream_t stream) {
  const float* x   = (const float*)d_in[0];   // [32,2048,256] f32
  const float* emb = (const float*)d_in[1];   // [4096,256]    f32
  float* out = (float*)d_out;                 // [NTOK*DIM] quantised + [1] loss
  float* ws  = (float*)d_ws;                  // [VOCAB] e2 + [NTOK/MB] partials

  vq_prep<<<VOCAB / 256, 256, 0, stream>>>(emb, ws);
  vq_main<<<NTOK / MB, 256, SMEM_BYTES, stream>>>(x, emb, ws, out, ws + VOCAB);
  vq_final<<<1, 64, 0, stream>>>(ws, out);
}